// ContextBranch_1245540516450
// MI455X (gfx1250) — compile-verified
//
#include <hip/hip_runtime.h>
#include <hip/hip_bf16.h>

// ---------------------------------------------------------------------------
// ContextBranch for MI455X (gfx1250, wave32).
//   features [256,64,64] f32, boxes [128,4] f32, w_fuse [256,2048] f32,
//   b_fuse [256] f32  ->  out [128,256,7,7] f32.
// Stage 1: ROIAlignV2 of 1024 context boxes, written straight into the GEMM
//          A-matrix layout A[m=6272][k=2048]  (m = (b/8)*49 + hw,
//          k = (b%8)*256 + c), matching the reference's reshape semantics.
// Stage 2: f32 WMMA GEMM (V_WMMA_F32_16X16X4_F32) + bias + ReLU.
// Workspace: 6272*2048*4 = 51,380,224 bytes in d_ws.
// ---------------------------------------------------------------------------

typedef float v2f __attribute__((ext_vector_type(2)));
typedef float v8f __attribute__((ext_vector_type(8)));

#define OUTSZ 7
#define KTOT  2048
#define MTOT  6272   // 128 * 49

__global__ void roi_pool_kernel(const float* __restrict__ feat,   // [256,64,64]
                                const float* __restrict__ boxes,  // [128,4]
                                float* __restrict__ A)            // [6272,2048]
{
    const int b = blockIdx.x;        // context-box index, 0..1023 (offset-major)
    const int c = threadIdx.x;       // channel, 0..255
    const int off = b >> 7;          // 0..7
    const int n   = b & 127;         // original box

    // 3x3 grid offsets skipping center, i-major order
    const int ii[8] = {0, 0, 0, 1, 1, 2, 2, 2};
    const int jj[8] = {0, 1, 2, 0, 2, 0, 1, 2};

    const float x1 = boxes[n * 4 + 0];
    const float y1 = boxes[n * 4 + 1];
    const float x2 = boxes[n * 4 + 2];
    const float y2 = boxes[n * 4 + 3];
    const float w3 = (x2 - x1) * (1.0f / 3.0f);
    const float h3 = (y2 - y1) * (1.0f / 3.0f);
    const float bx1 = x1 + jj[off] * w3;
    const float by1 = y1 + ii[off] * h3;
    const float bx2 = bx1 + w3;
    const float by2 = by1 + h3;

    const float SCALE = 1.0f / 16.0f;
    const float cx1 = bx1 * SCALE - 0.5f;
    const float cy1 = by1 * SCALE - 0.5f;
    const float bw  = (bx2 * SCALE - 0.5f - cx1) * (1.0f / OUTSZ);
    const float bh  = (by2 * SCALE - 0.5f - cy1) * (1.0f / OUTSZ);

    const int m_base = (b >> 3) * 49;          // (b/8)*49
    const int kcol   = (b & 7) * 256 + c;      // (b%8)*256 + c
    const float* fc  = feat + c * 4096;        // channel plane [64,64]

    for (int hw = 0; hw < 49; ++hw) {
        const int h = hw / 7;
        const int w = hw - h * 7;
        float acc = 0.0f;
        #pragma unroll
        for (int sy = 0; sy < 2; ++sy) {
            const float ty = (float)h + (sy + 0.5f) * 0.5f;
            const float y  = cy1 + ty * bh;
            const bool  vy = (y >= -1.0f) && (y <= 64.0f);
            float yc = fminf(fmaxf(y, 0.0f), 63.0f);
            float yl = fminf(floorf(yc), 62.0f);
            const int   yi = (int)yl;
            const float fy = yc - yl;
            #pragma unroll
            for (int sx = 0; sx < 2; ++sx) {
                const float tx = (float)w + (sx + 0.5f) * 0.5f;
                const float x  = cx1 + tx * bw;
                const bool  vx = (x >= -1.0f) && (x <= 64.0f);
                float xc = fminf(fmaxf(x, 0.0f), 63.0f);
                float xl = fminf(floorf(xc), 62.0f);
                const int   xi = (int)xl;
                const float fx = xc - xl;
                if (vy && vx) {
                    const int base = yi * 64 + xi;
                    const float f00 = fc[base];
                    const float f01 = fc[base + 1];
                    const float f10 = fc[base + 64];
                    const float f11 = fc[base + 65];
                    acc += f00 * (1.0f - fy) * (1.0f - fx)
                         + f01 * (1.0f - fy) * fx
                         + f10 * fy * (1.0f - fx)
                         + f11 * fy * fx;
                }
            }
        }
        // coalesced: consecutive c -> consecutive k
        A[(size_t)(m_base + hw) * KTOT + kcol] = acc * 0.25f;
    }
}

// One block = one 16-row M tile; 8 waves, each wave owns two 16-wide N tiles.
// A-frag / B-frag layout for V_WMMA_F32_16X16X4_F32 (ISA 7.12.2):
//   lane L holds row (L%16), K = kpair(+1) with kpair = 2*(L/16) -> float2 load.
__global__ void __launch_bounds__(256)
fuse_gemm_kernel(const float* __restrict__ A,     // [6272,2048]
                 const float* __restrict__ W,     // [256,2048]
                 const float* __restrict__ bias,  // [256]
                 float* __restrict__ out)         // [128,256,7,7]
{
    const int m0   = blockIdx.x * 16;
    const int wave = threadIdx.x >> 5;
    const int lane = threadIdx.x & 31;
    const int o0   = wave * 32;
    const int row  = lane & 15;
    const int kpair = (lane >> 4) << 1;   // 0 or 2

    const float* a_ptr  = A + (size_t)(m0 + row) * KTOT + kpair;
    const float* b_ptr0 = W + (size_t)(o0 + row) * KTOT + kpair;
    const float* b_ptr1 = W + (size_t)(o0 + 16 + row) * KTOT + kpair;

    v8f acc0 = {};
    v8f acc1 = {};

    for (int k0 = 0; k0 < KTOT; k0 += 4) {
        __builtin_prefetch(a_ptr + k0 + 512, 0, 1);   // global_prefetch_b8
        const v2f a  = *(const v2f*)(a_ptr  + k0);
        const v2f b0 = *(const v2f*)(b_ptr0 + k0);
        const v2f b1 = *(const v2f*)(b_ptr1 + k0);
        acc0 = __builtin_amdgcn_wmma_f32_16x16x4_f32(
                   false, a, false, b0, (short)0, acc0, false, false);
        acc1 = __builtin_amdgcn_wmma_f32_16x16x4_f32(
                   false, a, false, b1, (short)0, acc1, false, false);
    }

    // Epilogue: C/D 16x16 f32 layout -> VGPR r, lane L: M = r + 8*(L/16), N = L%16
    const float bias0 = bias[o0 + row];
    const float bias1 = bias[o0 + 16 + row];
    const int mbase = m0 + ((lane >> 4) << 3);
    #pragma unroll
    for (int r = 0; r < 8; ++r) {
        const int m  = mbase + r;            // < 6272 always (392*16 == 6272)
        const int nb = m / 49;
        const int hw = m - nb * 49;
        float v0 = acc0[r] + bias0;
        float v1 = acc1[r] + bias1;
        v0 = v0 > 0.0f ? v0 : 0.0f;
        v1 = v1 > 0.0f ? v1 : 0.0f;
        out[((size_t)nb * 256 + (o0 + row)) * 49 + hw]      = v0;
        out[((size_t)nb * 256 + (o0 + 16 + row)) * 49 + hw] = v1;
    }
}

extern "C" void kernel_launch(void* const* d_in, const int* in_sizes, int n_in,
                              void* d_out, int out_size, void* d_ws, size_t ws_size,
                              hipStream_t stream) {
    (void)in_sizes; (void)n_in; (void)out_size; (void)ws_size;
    const float* features = (const float*)d_in[0];  // [256,64,64]
    const float* boxes    = (const float*)d_in[1];  // [128,4]
    const float* w_fuse   = (const float*)d_in[2];  // [256,2048]
    const float* b_fuse   = (const float*)d_in[3];  // [256]
    float* out = (float*)d_out;                     // [128,256,7,7]
    float* A   = (float*)d_ws;                      // [6272,2048] f32, ~51.4 MB

    roi_pool_kernel<<<1024, 256, 0, stream>>>(features, boxes, A);
    fuse_gemm_kernel<<<MTOT / 16, 256, 0, stream>>>(A, w_fuse, b_fuse, out);
}